// AdaptiveEntropy_44040594653425
// MI455X (gfx1250) — compile-verified
//
#include <hip/hip_runtime.h>
#include <hip/hip_bf16.h>
#include <math.h>

// Problem constants (fixed by the reference):
//   x: (B=2, C=64, D=H=W=64) f32 ; w1: (8,64) ; w2: (1,8) ; BINS=128, POOL=4
#define NB   2
#define NC   64
#define NCM  8
#define NS   64          // 4*4*4 pooled spatial
#define BINS 128

typedef __attribute__((ext_vector_type(2))) float v2f;
typedef __attribute__((ext_vector_type(8))) float v8f;

// -------- workspace layout (4-byte words) --------
// [0,     8192)  pooled p  (B*C*64 floats)
// [8192,  8320)  s_small   (128 floats)
// [8320]         min key (uint, monotone float encoding)
// [8321]         max key (uint)
// [8322,  8450)  hist (128 uint)
#define WS_P    0
#define WS_S    8192
#define WS_MINK 8320
#define WS_MAXK 8321
#define WS_HIST 8322

// ---- monotone float<->uint key (total order preserved) ----
__device__ __forceinline__ unsigned f2key(float f) {
    unsigned b = __float_as_uint(f);
    return (b & 0x80000000u) ? ~b : (b | 0x80000000u);
}
__device__ __forceinline__ float key2f(unsigned k) {
    unsigned b = (k & 0x80000000u) ? (k & 0x7FFFFFFFu) : ~k;
    return __uint_as_float(b);
}

// ---- trilinear axis coeffs, align_corners=False, 4 -> 64 ----
__device__ __forceinline__ void axis_interp(int i, int& lo, int& hi, float& f) {
    float c = ((float)i + 0.5f) * 0.0625f - 0.5f;   // (i+0.5)*4/64 - 0.5
    c = fminf(fmaxf(c, 0.0f), 3.0f);
    float fl = floorf(c);
    lo = (int)fl;
    hi = lo + 1; if (hi > 3) hi = 3;
    f = c - fl;
}

__device__ __forceinline__ float tri_s(const float* __restrict__ sl, int b,
                                       int ld, int hd, float fd,
                                       int lh, int hh, float fh, int w) {
    int lw, hw; float fw;
    axis_interp(w, lw, hw, fw);
    const float* sb = sl + b * 64;
    float c00 = sb[ld*16 + lh*4 + lw] * (1.f - fw) + sb[ld*16 + lh*4 + hw] * fw;
    float c01 = sb[ld*16 + hh*4 + lw] * (1.f - fw) + sb[ld*16 + hh*4 + hw] * fw;
    float c10 = sb[hd*16 + lh*4 + lw] * (1.f - fw) + sb[hd*16 + lh*4 + hw] * fw;
    float c11 = sb[hd*16 + hh*4 + lw] * (1.f - fw) + sb[hd*16 + hh*4 + hw] * fw;
    float c0 = c00 * (1.f - fh) + c01 * fh;
    float c1 = c10 * (1.f - fh) + c11 * fh;
    return c0 * (1.f - fd) + c1 * fd;
}

// =====================================================================
// Kernel 1: adaptive avg pool 64^3 -> 4^3. One block per (b,c,pd,ph) slab
// (16 d-planes x 16 h-rows x 64 w = 16384 floats, fully coalesced b128).
// =====================================================================
__global__ void k_pool(const float* __restrict__ x, float* __restrict__ ws_f) {
    int blk = blockIdx.x;
    int ph = blk & 3, pd = (blk >> 2) & 3, c = (blk >> 4) & 63, b = blk >> 10;
    const float* base = x + (((size_t)((b * NC + c) * 64 + pd * 16) * 64 + ph * 16) * 64);
    int t = threadIdx.x;          // 256 threads; each loads one float4 per d-plane
    float acc = 0.f;
#pragma unroll
    for (int dd = 0; dd < 16; ++dd) {
        const float4 v = *reinterpret_cast<const float4*>(base + (size_t)dd * 4096 + t * 4);
        if (dd < 15)
            __builtin_prefetch(base + (size_t)(dd + 1) * 4096 + t * 4, 0, 1); // global_prefetch_b8
        acc += (v.x + v.y) + (v.z + v.w);
    }
    // thread t -> (h = t/16, w4 = t%16); pw = (t%16)/4; group index g in [0,64)
    __shared__ float red[256];
    int pw = (t & 15) >> 2;
    int g  = ((t >> 4) << 2) | (t & 3);
    red[pw * 64 + g] = acc;
    for (int st = 32; st > 0; st >>= 1) {
        __syncthreads();
        if (g < st) red[pw * 64 + g] += red[pw * 64 + g + st];
    }
    __syncthreads();
    if (t < 4) {
        int s = pd * 16 + ph * 4 + t;
        ws_f[WS_P + ((b * NC + c) * NS + s)] = red[t * 64] * (1.0f / 4096.0f);
    }
}

// =====================================================================
// Kernel 2: channel squeeze via V_WMMA_F32_16X16X4_F32 (exact f32 MMA),
// instance-norm, exact-erf GELU, w2 contraction, sigmoid -> s_small[128].
// Also (re)initializes min/max keys and the histogram each call.
// One block, 128 threads = 4 waves; each wave owns 2 N-tiles of 16.
//   D(16x16) += A(16x4: padded w1) * B(4x16: p columns), 16 k-steps, K=64.
// =====================================================================
__global__ void k_squeeze(const float* __restrict__ w1, const float* __restrict__ w2,
                          float* __restrict__ ws_f, unsigned* __restrict__ ws_u) {
    __shared__ float p_l[NB * NC * NS];   // 8192
    __shared__ float w1p[16 * 64];        // w1 padded M=8 -> 16
    __shared__ float w2_l[8];
    __shared__ float h_l[8 * 128];        // h[o][b*64+s]
    __shared__ float mu_l[16], rs_l[16];

    int t = threadIdx.x;                  // 128 threads
    for (int i = t; i < NB * NC * NS; i += 128) p_l[i] = ws_f[WS_P + i];
    for (int i = t; i < 16 * 64; i += 128) {
        int o = i >> 6, c = i & 63;
        w1p[i] = (o < NCM) ? w1[o * 64 + c] : 0.0f;
    }
    if (t < 8) w2_l[t] = w2[t];
    ws_u[WS_HIST + t] = 0u;               // zero histogram for kernel 4
    if (t == 0) ws_u[WS_MINK] = 0xFFFFFFFFu;
    if (t == 1) ws_u[WS_MAXK] = 0u;
    __syncthreads();

    int wave = t >> 5, lane = t & 31;
    int half = lane >> 4, m = lane & 15;  // A: lanes 0-15 hold K=0,1 ; 16-31 hold K=2,3
    for (int tile = wave * 2; tile < wave * 2 + 2; ++tile) {
        int nb = tile * 16;               // global column base (N = b*64+s, 128 total)
        int bb = nb >> 6;
        int sb = (nb & 63) + m;           // column within batch bb
        v8f acc = {};
#pragma unroll
        for (int k0 = 0; k0 < 64; k0 += 4) {
            int k = k0 + 2 * half;
            v2f a, bv;
            a.x = w1p[m * 64 + k];
            a.y = w1p[m * 64 + k + 1];
            bv.x = p_l[(bb * NC + k) * NS + sb];
            bv.y = p_l[(bb * NC + k + 1) * NS + sb];
            acc = __builtin_amdgcn_wmma_f32_16x16x4_f32(
                false, a, false, bv, (short)0, acc, false, false);
        }
        // C/D layout: VGPR r = rows M=r (lanes 0-15) / M=r+8 (lanes 16-31).
        // Only M=0..7 are real outputs -> lanes 0-15 carry everything we need.
        if (lane < 16) {
#pragma unroll
            for (int r = 0; r < 8; ++r) h_l[r * 128 + nb + lane] = acc[r];
        }
    }
    __syncthreads();

    if (t < 16) {                         // instance norm per (b, o)
        int bb = t >> 3, o = t & 7;
        const float* row = &h_l[o * 128 + bb * 64];
        float s = 0.f, s2 = 0.f;
        for (int i = 0; i < 64; ++i) { float v = row[i]; s += v; s2 += v * v; }
        float mean = s * (1.f / 64.f);
        float var  = s2 * (1.f / 64.f) - mean * mean;
        mu_l[t] = mean;
        rs_l[t] = rsqrtf(var + 1e-5f);
    }
    __syncthreads();

    {                                     // t = b*64 + spatial
        int bb = t >> 6;
        float acc2 = 0.f;
#pragma unroll
        for (int o = 0; o < 8; ++o) {
            float v  = (h_l[o * 128 + t] - mu_l[bb * 8 + o]) * rs_l[bb * 8 + o];
            float gl = 0.5f * v * (1.0f + erff(v * 0.70710678118f)); // exact GELU
            acc2 += w2_l[o] * gl;
        }
        ws_f[WS_S + t] = 1.0f / (1.0f + expf(-acc2));
    }
}

// =====================================================================
// Kernel 3: weighted = x * upsample(s); global min/max via monotone keys.
// =====================================================================
__global__ void k_minmax(const float* __restrict__ x, const float* __restrict__ ws_f,
                         unsigned* __restrict__ ws_u) {
    __shared__ float sl[128];
    int t = threadIdx.x;
    if (t < 128) sl[t] = ws_f[WS_S + t];
    __syncthreads();

    unsigned kmin = 0xFFFFFFFFu, kmax = 0u;
    const unsigned N4 = 8388608u;         // 2*64*64^3 / 4
    unsigned stride = gridDim.x * blockDim.x;
    for (unsigned i4 = blockIdx.x * blockDim.x + t; i4 < N4; i4 += stride) {
        unsigned e = i4 << 2;
        int w = e & 63, h = (e >> 6) & 63, d = (e >> 12) & 63, b = e >> 24;
        int ld, hd, lh, hh; float fd, fh;
        axis_interp(d, ld, hd, fd);
        axis_interp(h, lh, hh, fh);
        float4 xv = reinterpret_cast<const float4*>(x)[i4];
        float v0 = xv.x * tri_s(sl, b, ld, hd, fd, lh, hh, fh, w + 0);
        float v1 = xv.y * tri_s(sl, b, ld, hd, fd, lh, hh, fh, w + 1);
        float v2 = xv.z * tri_s(sl, b, ld, hd, fd, lh, hh, fh, w + 2);
        float v3 = xv.w * tri_s(sl, b, ld, hd, fd, lh, hh, fh, w + 3);
        unsigned k0 = f2key(v0), k1 = f2key(v1), k2 = f2key(v2), k3 = f2key(v3);
        unsigned a = (k0 < k1) ? k0 : k1, bq = (k2 < k3) ? k2 : k3;
        unsigned lo = (a < bq) ? a : bq;
        a = (k0 > k1) ? k0 : k1; bq = (k2 > k3) ? k2 : k3;
        unsigned hi = (a > bq) ? a : bq;
        kmin = (lo < kmin) ? lo : kmin;
        kmax = (hi > kmax) ? hi : kmax;
    }
    __shared__ unsigned rmin[256], rmax[256];
    rmin[t] = kmin; rmax[t] = kmax;
    for (int st = 128; st > 0; st >>= 1) {
        __syncthreads();
        if (t < st) {
            unsigned a = rmin[t + st]; if (a < rmin[t]) rmin[t] = a;
            unsigned b2 = rmax[t + st]; if (b2 > rmax[t]) rmax[t] = b2;
        }
    }
    if (t == 0) {
        atomicMin(&ws_u[WS_MINK], rmin[0]);
        atomicMax(&ws_u[WS_MAXK], rmax[0]);
    }
}

// =====================================================================
// Kernel 4: second streaming pass, LDS-privatized 128-bin histogram.
// =====================================================================
__global__ void k_hist(const float* __restrict__ x, const float* __restrict__ ws_f,
                       unsigned* __restrict__ ws_u) {
    __shared__ float sl[128];
    __shared__ unsigned hl[128];
    int t = threadIdx.x;
    if (t < 128) { sl[t] = ws_f[WS_S + t]; hl[t] = 0u; }
    __syncthreads();

    float mn = key2f(ws_u[WS_MINK]);
    float mx = key2f(ws_u[WS_MAXK]);
    float inv = 1.0f / (mx - mn + 1e-8f);

    const unsigned N4 = 8388608u;
    unsigned stride = gridDim.x * blockDim.x;
    for (unsigned i4 = blockIdx.x * blockDim.x + t; i4 < N4; i4 += stride) {
        unsigned e = i4 << 2;
        int w = e & 63, h = (e >> 6) & 63, d = (e >> 12) & 63, b = e >> 24;
        int ld, hd, lh, hh; float fd, fh;
        axis_interp(d, ld, hd, fd);
        axis_interp(h, lh, hh, fh);
        float4 xv = reinterpret_cast<const float4*>(x)[i4];
        float v[4];
        v[0] = xv.x * tri_s(sl, b, ld, hd, fd, lh, hh, fh, w + 0);
        v[1] = xv.y * tri_s(sl, b, ld, hd, fd, lh, hh, fh, w + 1);
        v[2] = xv.z * tri_s(sl, b, ld, hd, fd, lh, hh, fh, w + 2);
        v[3] = xv.w * tri_s(sl, b, ld, hd, fd, lh, hh, fh, w + 3);
#pragma unroll
        for (int j = 0; j < 4; ++j) {
            float xn = (v[j] - mn) * inv;
            int idx = (int)floorf(xn * (float)BINS);
            idx = (idx < 0) ? 0 : ((idx > BINS - 1) ? BINS - 1 : idx);
            atomicAdd(&hl[idx], 1u);
        }
    }
    __syncthreads();
    if (t < 128 && hl[t]) atomicAdd(&ws_u[WS_HIST + t], hl[t]);
}

// =====================================================================
// Kernel 5: entropy of histogram -> d_out[0].
// =====================================================================
__global__ void k_entropy(const unsigned* __restrict__ ws_u, float* __restrict__ out) {
    __shared__ float red[128];
    int t = threadIdx.x;                  // 128 threads
    float cnt = (float)ws_u[WS_HIST + t];
    red[t] = cnt;
    for (int st = 64; st > 0; st >>= 1) {
        __syncthreads();
        if (t < st) red[t] += red[t + st];
    }
    __syncthreads();
    float total = red[0];
    __syncthreads();
    float prob = cnt / (total + 1e-10f);
    red[t] = prob * log2f(prob + 1e-10f);
    for (int st = 64; st > 0; st >>= 1) {
        __syncthreads();
        if (t < st) red[t] += red[t + st];
    }
    __syncthreads();
    if (t == 0) out[0] = -red[0];
}

extern "C" void kernel_launch(void* const* d_in, const int* in_sizes, int n_in,
                              void* d_out, int out_size, void* d_ws, size_t ws_size,
                              hipStream_t stream) {
    const float* x  = (const float*)d_in[0];
    const float* w1 = (const float*)d_in[1];
    const float* w2 = (const float*)d_in[2];
    float*    out  = (float*)d_out;
    float*    ws_f = (float*)d_ws;
    unsigned* ws_u = (unsigned*)d_ws;

    k_pool   <<<2048, 256, 0, stream>>>(x, ws_f);
    k_squeeze<<<1,    128, 0, stream>>>(w1, w2, ws_f, ws_u);
    k_minmax <<<4096, 256, 0, stream>>>(x, ws_f, ws_u);
    k_hist   <<<4096, 256, 0, stream>>>(x, ws_f, ws_u);
    k_entropy<<<1,    128, 0, stream>>>(ws_u, out);
}